// PolarConvLayer_36928128811345
// MI455X (gfx1250) — compile-verified
//
#include <hip/hip_runtime.h>

#define D_FEAT 128
#define WAVE 32

// ---------------------------------------------------------------------------
// Hardware fp32 atomic add (global_atomic_add_f32, no return -> STOREcnt)
// ---------------------------------------------------------------------------
__device__ __forceinline__ void atom_add_f32(float* p, float v) {
    unsafeAtomicAdd(p, v);
}

// Uniform lane broadcast via v_readlane_b32 (result lands in an SGPR).
__device__ __forceinline__ int bcast_i32(int v, int lane) {
    return __builtin_amdgcn_readlane(v, lane);
}
__device__ __forceinline__ float bcast_f32(float v, int lane) {
    return __uint_as_float(__builtin_amdgcn_readlane(__float_as_uint(v), lane));
}

// ---------------------------------------------------------------------------
// Kernel 1: zero the output (and degree scratch if GCN mode)
// ---------------------------------------------------------------------------
__global__ __launch_bounds__(256)
void zero_init_kernel(float4* __restrict__ out, int n4,
                      float* __restrict__ deg, int ndeg,
                      const int* __restrict__ mode) {
    int tid    = blockIdx.x * blockDim.x + threadIdx.x;
    int stride = gridDim.x * blockDim.x;
    float4 z = make_float4(0.f, 0.f, 0.f, 0.f);
    for (int i = tid; i < n4; i += stride) out[i] = z;
    if (mode[0] != 1) {
        for (int i = tid; i < ndeg; i += stride) deg[i] = 0.f;
    }
}

// ---------------------------------------------------------------------------
// Kernel 2 (mode==0 only): out-degree via hardware f32 atomics
// ---------------------------------------------------------------------------
__global__ __launch_bounds__(256)
void degree_kernel(const int* __restrict__ src, float* __restrict__ deg,
                   const int* __restrict__ mode, int E) {
    if (mode[0] == 1) return;
    int tid    = blockIdx.x * blockDim.x + threadIdx.x;
    int stride = gridDim.x * blockDim.x;
    for (int i = tid; i < E; i += stride)
        atom_add_f32(&deg[src[i]], 1.0f);
}

// ---------------------------------------------------------------------------
// Kernel 3: the scatter-add SpMM.
// One wave32 processes a batch of 32 edges:
//   - lane i loads metadata (src/dst/weight) of edge base+i  (coalesced)
//   - lane i prefetches all 4 cachelines of h[src[base+i]]   (global_prefetch_b8)
//   - loop j=0..31: broadcast edge j's metadata to SGPRs via v_readlane,
//     wave gathers the full 512B row (global_load_b128, scalar base + lane
//     offset), scales, and scatters with 4 no-return global_atomic_add_f32.
// h and out both fit in the 192MB L2, so gathers/atomics run at L2 speed.
// Zero LDS used -> full 320KB LDS/WGP$ pool available as cache.
// ---------------------------------------------------------------------------
__global__ __launch_bounds__(256)
void scatter_spmm_kernel(const float* __restrict__ h,
                         const float* __restrict__ w,
                         const int* __restrict__ src,
                         const int* __restrict__ dst,
                         const int* __restrict__ mode,
                         const float* __restrict__ deg,
                         float* __restrict__ out,
                         int E) {
    const int lane   = threadIdx.x & (WAVE - 1);
    const int waveIb = threadIdx.x >> 5;
    const long wid   = (long)blockIdx.x * (blockDim.x >> 5) + waveIb;
    const long base  = wid * WAVE;
    if (base >= E) return;

    const int myMode = mode[0];

    // ---- per-lane edge metadata (edge base+lane) ----
    long e  = base + lane;
    int  s  = 0, d = 0;
    float wt = 0.f;
    if (e < E) {
        s = src[e];
        d = dst[e];
        wt = (myMode == 1) ? w[e] : rsqrtf(deg[s]);
    }

    // ---- prefetch this lane's source row (4 x 128B cachelines) ----
    const char* hrow = (const char*)(h + (size_t)s * D_FEAT);
    __builtin_prefetch(hrow +   0, 0, 0);
    __builtin_prefetch(hrow + 128, 0, 0);
    __builtin_prefetch(hrow + 256, 0, 0);
    __builtin_prefetch(hrow + 384, 0, 0);

    if (base + WAVE <= E) {
        // fast path: full batch, no bounds checks
        #pragma unroll
        for (int j = 0; j < WAVE; ++j) {
            int   sj = bcast_i32(s,  j);   // SGPR
            int   dj = bcast_i32(d,  j);   // SGPR
            float wj = bcast_f32(wt, j);   // SGPR
            const float4* hp = (const float4*)(h + (size_t)sj * D_FEAT) + lane;
            float4 v = *hp;
            float* op = out + (size_t)dj * D_FEAT + lane * 4;
            atom_add_f32(op + 0, v.x * wj);
            atom_add_f32(op + 1, v.y * wj);
            atom_add_f32(op + 2, v.z * wj);
            atom_add_f32(op + 3, v.w * wj);
        }
    } else {
        // tail batch
        for (int j = 0; j < WAVE; ++j) {
            if (base + j >= E) break;          // wave-uniform
            int   sj = bcast_i32(s,  j);
            int   dj = bcast_i32(d,  j);
            float wj = bcast_f32(wt, j);
            const float4* hp = (const float4*)(h + (size_t)sj * D_FEAT) + lane;
            float4 v = *hp;
            float* op = out + (size_t)dj * D_FEAT + lane * 4;
            atom_add_f32(op + 0, v.x * wj);
            atom_add_f32(op + 1, v.y * wj);
            atom_add_f32(op + 2, v.z * wj);
            atom_add_f32(op + 3, v.w * wj);
        }
    }
}

// ---------------------------------------------------------------------------
// Kernel 4 (mode==0 only): out[n,:] *= rsqrt(deg[n])
// ---------------------------------------------------------------------------
__global__ __launch_bounds__(256)
void finalize_kernel(float* __restrict__ out, const float* __restrict__ deg,
                     const int* __restrict__ mode, int N) {
    if (mode[0] == 1) return;
    int tid    = blockIdx.x * blockDim.x + threadIdx.x;
    int stride = gridDim.x * blockDim.x;
    int total  = N * (D_FEAT / 4);
    for (int i = tid; i < total; i += stride) {
        int node = i / (D_FEAT / 4);
        int c    = i % (D_FEAT / 4);
        float sc = rsqrtf(deg[node]);
        float4* p = (float4*)(out + (size_t)node * D_FEAT) + c;
        float4 v = *p;
        v.x *= sc; v.y *= sc; v.z *= sc; v.w *= sc;
        *p = v;
    }
}

// ---------------------------------------------------------------------------
// Launcher
// Inputs (setup_inputs order): h [N*128] f32, w [E] f32, src [E] i32,
//                              dst [E] i32, mode [1] i32
// ---------------------------------------------------------------------------
extern "C" void kernel_launch(void* const* d_in, const int* in_sizes, int n_in,
                              void* d_out, int out_size, void* d_ws, size_t ws_size,
                              hipStream_t stream) {
    const float* h    = (const float*)d_in[0];
    const float* w    = (const float*)d_in[1];
    const int*   src  = (const int*)d_in[2];
    const int*   dst  = (const int*)d_in[3];
    const int*   mode = (const int*)d_in[4];
    float*       out  = (float*)d_out;
    float*       deg  = (float*)d_ws;   // N floats of scratch (mode==0 only)

    const int N = in_sizes[0] / D_FEAT;
    const int E = in_sizes[2];
    const int n4 = out_size / 4;
    int degCap = (int)(ws_size / sizeof(float));
    if (degCap > N) degCap = N;

    // 1) zero output (+ degree scratch when in GCN mode)
    zero_init_kernel<<<1024, 256, 0, stream>>>((float4*)out, n4, deg, degCap, mode);

    // 2) degree accumulation (early-exits when mode==1)
    {
        int blocks = (E + 255) / 256;
        if (blocks > 4096) blocks = 4096;
        degree_kernel<<<blocks, 256, 0, stream>>>(src, deg, mode, E);
    }

    // 3) main scatter-add SpMM: one wave32 per 32-edge batch, 8 waves/block
    {
        long waves  = ((long)E + WAVE - 1) / WAVE;
        long blocks = (waves + 7) / 8;
        scatter_spmm_kernel<<<(int)blocks, 256, 0, stream>>>(h, w, src, dst,
                                                             mode, deg, out, E);
    }

    // 4) final symmetric normalization (early-exits when mode==1)
    {
        int total  = N * (D_FEAT / 4);
        int blocks = (total + 255) / 256;
        if (blocks > 4096) blocks = 4096;
        finalize_kernel<<<blocks, 256, 0, stream>>>(out, deg, mode, N);
    }
}